// VisibleRebuildModule_2602750181487
// MI455X (gfx1250) — compile-verified
//
#include <hip/hip_runtime.h>
#include <hip/hip_bf16.h>
#include <math.h>

#define HW 256
#define NPTS (HW * HW)
#define NB 128
#define NKPT 68
#define OFFSET_SCALE 6.0f

typedef float v2f __attribute__((ext_vector_type(2)));
typedef float v8f __attribute__((ext_vector_type(8)));

// ---------------------------------------------------------------------------
// Kernel 1: per-batch similarity transform solve (Kabsch via Newton polar).
// One block per batch. Threads 0..67 gather keypoints to LDS; wave 0 reduces
// with __shfl_xor (wave32) and lane 0 runs the 3x3 polar iteration
// (R = V U^T is exactly the orthogonal polar factor of H^T; the det-scaled
// Newton iteration Y <- 0.5*(mu*Y + (mu*Y)^-1) converges to polar(H) = U V^T,
// then R = polar(H)^T — matches the reference's svd-derived R, including the
// uncorrected reflection case).
// Writes 12 floats per batch to ws: R*s (row major, 9) then t (3).
// ---------------------------------------------------------------------------
__global__ __launch_bounds__(128) void solve_tform_kernel(
    const float* __restrict__ off, const float* __restrict__ kpt,
    const float* __restrict__ meanp, const int* __restrict__ uv,
    float* __restrict__ Rt) {
  const int b = blockIdx.x;
  const int tid = threadIdx.x;

  __shared__ float s_src[NKPT][3];
  __shared__ float s_dst[NKPT][3];

  if (tid < NKPT) {
    const int u = uv[2 * tid + 0];
    const int v = uv[2 * tid + 1];
    const int sp = u * HW + v;
#pragma unroll
    for (int c = 0; c < 3; ++c) {
      s_src[tid][c] = off[((size_t)(b * 3 + c) << 16) + sp] * OFFSET_SCALE +
                      meanp[((size_t)c << 16) + sp];
      s_dst[tid][c] = kpt[((size_t)(b * 3 + c) << 16) + sp];
    }
  }
  __syncthreads();
  if (tid >= 32) return;  // one wave does the math; no barriers below

  const float r0s0 = s_src[0][0], r0s1 = s_src[0][1], r0s2 = s_src[0][2];
  const float r0d0 = s_dst[0][0], r0d1 = s_dst[0][1], r0d2 = s_dst[0][2];

  // Pass 1: sums for means and the two distance sums.
  float ss0 = 0.f, ss1 = 0.f, ss2 = 0.f, sd0 = 0.f, sd1 = 0.f, sd2 = 0.f;
  float dist_s = 0.f, dist_d = 0.f;
  for (int k = tid; k < NKPT; k += 32) {
    const float a0 = s_src[k][0], a1 = s_src[k][1], a2 = s_src[k][2];
    const float b0 = s_dst[k][0], b1 = s_dst[k][1], b2 = s_dst[k][2];
    ss0 += a0; ss1 += a1; ss2 += a2;
    sd0 += b0; sd1 += b1; sd2 += b2;
    if (k > 0) {
      const float dx = a0 - r0s0, dy = a1 - r0s1, dz = a2 - r0s2;
      dist_s += sqrtf(dx * dx + dy * dy + dz * dz);
      const float ex = b0 - r0d0, ey = b1 - r0d1, ez = b2 - r0d2;
      dist_d += sqrtf(ex * ex + ey * ey + ez * ez);
    }
  }
#define WRED(x) do { for (int o = 16; o > 0; o >>= 1) x += __shfl_xor(x, o, 32); } while (0)
  WRED(ss0); WRED(ss1); WRED(ss2);
  WRED(sd0); WRED(sd1); WRED(sd2);
  WRED(dist_s); WRED(dist_d);

  const float s = dist_d / dist_s;
  const float inv_n = 1.0f / (float)NKPT;
  const float ms0 = ss0 * inv_n, ms1 = ss1 * inv_n, ms2 = ss2 * inv_n;
  const float md0 = sd0 * inv_n, md1 = sd1 * inv_n, md2 = sd2 * inv_n;

  // Pass 2: H = AA^T @ BB, AA_k = (src_k - mean_src)*s, BB_k = dst_k - mean_dst.
  float H[9] = {0.f, 0.f, 0.f, 0.f, 0.f, 0.f, 0.f, 0.f, 0.f};
  for (int k = tid; k < NKPT; k += 32) {
    const float ax = (s_src[k][0] - ms0) * s;
    const float ay = (s_src[k][1] - ms1) * s;
    const float az = (s_src[k][2] - ms2) * s;
    const float bx = s_dst[k][0] - md0;
    const float by = s_dst[k][1] - md1;
    const float bz = s_dst[k][2] - md2;
    H[0] += ax * bx; H[1] += ax * by; H[2] += ax * bz;
    H[3] += ay * bx; H[4] += ay * by; H[5] += ay * bz;
    H[6] += az * bx; H[7] += az * by; H[8] += az * bz;
  }
#pragma unroll
  for (int i = 0; i < 9; ++i) WRED(H[i]);
#undef WRED

  if (tid != 0) return;

  // Newton polar iteration on Y (init = H / ||H||_F).
  float Y[9];
  float fn = 0.f;
#pragma unroll
  for (int i = 0; i < 9; ++i) fn += H[i] * H[i];
  const float inv_fn = rsqrtf(fmaxf(fn, 1e-30f));
#pragma unroll
  for (int i = 0; i < 9; ++i) Y[i] = H[i] * inv_fn;

  for (int it = 0; it < 12; ++it) {
    // Row-major cofactor matrix C: Y^-T = C / det(Y).
    const float C0 = Y[4] * Y[8] - Y[5] * Y[7];
    const float C1 = Y[5] * Y[6] - Y[3] * Y[8];
    const float C2 = Y[3] * Y[7] - Y[4] * Y[6];
    const float C3 = Y[2] * Y[7] - Y[1] * Y[8];
    const float C4 = Y[0] * Y[8] - Y[2] * Y[6];
    const float C5 = Y[1] * Y[6] - Y[0] * Y[7];
    const float C6 = Y[1] * Y[5] - Y[2] * Y[4];
    const float C7 = Y[2] * Y[3] - Y[0] * Y[5];
    const float C8 = Y[0] * Y[4] - Y[1] * Y[3];
    float det = Y[0] * C0 + Y[1] * C1 + Y[2] * C2;
    if (fabsf(det) < 1e-30f) det = 1e-30f;
    const float mu = powf(fabsf(det), -1.0f / 3.0f);  // det-scaled Newton
    const float w0 = 0.5f * mu;
    const float w1 = 0.5f / (det * mu);
    Y[0] = w0 * Y[0] + w1 * C0; Y[1] = w0 * Y[1] + w1 * C3; Y[2] = w0 * Y[2] + w1 * C6;
    Y[3] = w0 * Y[3] + w1 * C1; Y[4] = w0 * Y[4] + w1 * C4; Y[5] = w0 * Y[5] + w1 * C7;
    Y[6] = w0 * Y[6] + w1 * C2; Y[7] = w0 * Y[7] + w1 * C5; Y[8] = w0 * Y[8] + w1 * C8;
  }
  // R = Y^T (3x3 orthogonal, same det sign as H).
  float R[9];
  R[0] = Y[0]; R[1] = Y[3]; R[2] = Y[6];
  R[3] = Y[1]; R[4] = Y[4]; R[5] = Y[7];
  R[6] = Y[2]; R[7] = Y[5]; R[8] = Y[8];

  // t = mean(dst) - s * R @ mean(src)
  const float t0 = md0 - s * (R[0] * ms0 + R[1] * ms1 + R[2] * ms2);
  const float t1 = md1 - s * (R[3] * ms0 + R[4] * ms1 + R[5] * ms2);
  const float t2 = md2 - s * (R[6] * ms0 + R[7] * ms1 + R[8] * ms2);

  float* o = Rt + b * 12;
#pragma unroll
  for (int i = 0; i < 9; ++i) o[i] = R[i] * s;
  o[9] = t0; o[10] = t1; o[11] = t2;
}

// ---------------------------------------------------------------------------
// Kernel 2: streaming affine transform via V_WMMA_F32_16X16X4_F32.
//   A (16x4): rows 0..2 = [R*s | t]   (built once per wave)
//   B (4x16): 16 homogeneous points (x,y,z,1) as columns
//   C rows 0..2 -> VGPR0..2, lanes 0..15 = point index -> direct planar store.
// f32 A/B layout: lanes 0-15 hold K=0,1 ; lanes 16-31 hold K=2,3 (per ISA).
// 512 blocks x 256 threads = 4096 waves = 32 waves/batch x 128 tiles of 16 pts.
//
// Branchless B assembly: upper-half lanes load the same y cacheline as the
// lower half (coalesces to one request) and v_cndmask selects 1.0, so EXEC
// stays all-ones through the whole load+WMMA stream (ISA requirement) with no
// per-iteration saveexec churn; only the (required) store predicate diverges.
// Cache policy: Offset is read once and the output written once (~200 MB
// streamed), while mean_posmap (768 KB) is re-read 128x -> NT loads/stores
// for the streams keep mean_posmap + Rt resident in the 192 MB L2.
// ---------------------------------------------------------------------------
__global__ __launch_bounds__(256) void transform_wmma_kernel(
    const float* __restrict__ off, const float* __restrict__ meanp,
    const float* __restrict__ Rt, float* __restrict__ out) {
  const int lane = threadIdx.x & 31;
  const int W = blockIdx.x * 8 + (threadIdx.x >> 5);  // global wave id, 0..4095
  const int b = W >> 5;                               // 32 waves per batch
  const int widx = W & 31;
  const int n = lane & 15;          // point column within tile
  const bool hi = (lane >> 4) != 0; // upper half-wave: holds K=2,3 of A/B

  // Build A = [R*s | t] rows 0..2 (rows 3..15 zero).
  const float* R = Rt + b * 12;
  v2f A;
  A.x = 0.f;
  A.y = 0.f;
  if (n < 3) {
    A.x = hi ? R[n * 3 + 2] : R[n * 3 + 0];
    A.y = hi ? R[9 + n]     : R[n * 3 + 1];
  }

  const int c0 = hi ? 2 : 0;  // this lane's B.x plane: x (lo) or z (hi)
  const float* src0 = off + ((size_t)(b * 3 + c0) << 16);
  const float* src1 = off + ((size_t)(b * 3 + 1) << 16);  // y plane
  const float* m0 = meanp + ((size_t)c0 << 16);
  const float* m1 = meanp + ((size_t)1 << 16);
  float* dst0 = out + ((size_t)(b * 3 + 0) << 16);
  float* dst1 = out + ((size_t)(b * 3 + 1) << 16);
  float* dst2 = out + ((size_t)(b * 3 + 2) << 16);

  const int pbase = widx * 2048 + n;
#pragma unroll 8
  for (int i = 0; i < 128; ++i) {
    const int p = pbase + i * 16;
    // Streaming (read-once) operand: non-temporal loads.
    const float o0 = __builtin_nontemporal_load(src0 + p);
    const float o1 = __builtin_nontemporal_load(src1 + p);
    // L2-resident operand: regular temporal loads.
    const float mm0 = m0[p];
    const float mm1 = m1[p];
    v2f Bv;
    Bv.x = fmaf(o0, OFFSET_SCALE, mm0);                     // x (lo) / z (hi)
    const float yv = fmaf(o1, OFFSET_SCALE, mm1);
    Bv.y = hi ? 1.0f : yv;                                  // y (lo) / 1 (hi)
    v8f C = {};
    C = __builtin_amdgcn_wmma_f32_16x16x4_f32(
        /*neg_a=*/false, A, /*neg_b=*/false, Bv,
        /*c_mod=*/(short)0, C, /*reuse_a=*/false, /*reuse_b=*/false);
    if (!hi) {  // rows 0..2 of C live in lanes 0..15 of VGPRs 0..2
      __builtin_nontemporal_store(C[0], dst0 + p);
      __builtin_nontemporal_store(C[1], dst1 + p);
      __builtin_nontemporal_store(C[2], dst2 + p);
    }
  }
}

extern "C" void kernel_launch(void* const* d_in, const int* in_sizes, int n_in,
                              void* d_out, int out_size, void* d_ws, size_t ws_size,
                              hipStream_t stream) {
  const float* Offset = (const float*)d_in[0];      // (128, 3, 256, 256)
  const float* Posmap_kpt = (const float*)d_in[1];  // (128, 3, 256, 256)
  const float* mean_posmap = (const float*)d_in[2]; // (3, 256, 256)
  const int* uv_kpt = (const int*)d_in[3];          // (68, 2)
  float* Rt = (float*)d_ws;                         // 128 * 12 floats
  float* outp = (float*)d_out;

  solve_tform_kernel<<<NB, 128, 0, stream>>>(Offset, Posmap_kpt, mean_posmap,
                                             uv_kpt, Rt);
  transform_wmma_kernel<<<512, 256, 0, stream>>>(Offset, mean_posmap, Rt, outp);
}